// ReservoirAttention_64707977282125
// MI455X (gfx1250) — compile-verified
//
#include <hip/hip_runtime.h>
#include <math.h>

#define RES   4096
#define EDIM  512
#define HEADS 8
#define DHD   64
#define SEQ   512
#define BATCH 32
#define ALEAK 0.3f
#define TCH   64      // time-chunk for WinU precompute
#define NCHUNK 256    // chunks for deterministic CSR build

typedef float v8f __attribute__((ext_vector_type(8)));
typedef float v2f __attribute__((ext_vector_type(2)));

// ---------------------------------------------------------------------------
// C[m][n] = (use_bias ? Bm[n][0] : 0) + sum_k A[m][k] * Bm[n][kcol0+k]
// Block: 256 thr = 8 waves; block tile 128(M) x 64(N); K staged via LDS, 32/chunk.
// Each wave: one 16(M) x 64(N) strip = 4 independent 16x16 WMMA accumulators,
// so consecutive v_wmma ops never share a D->C chain and the A fragment is
// reused across 4 matrix ops per k-step.
// Frag layout (ISA 7.12.2): A 16x4: lanes0-15 M=lane K={0,1}, lanes16-31 K={2,3}
//                           B 4x16 mirrored; C/D: vgpr v: M=v (+8 for hi lanes)
// ---------------------------------------------------------------------------
__global__ __launch_bounds__(256) void gemm_bt_wmma(
    const float* __restrict__ A, int lda,
    const float* __restrict__ Bm, int ldb, int kcol0, int use_bias,
    float* __restrict__ C, int ldc, int K)
{
    __shared__ float sA[128 * 36];
    __shared__ float sB[64 * 36];
    const int tid  = threadIdx.x;
    const int lane = tid & 31;
    const int wv   = tid >> 5;
    const int ml   = lane & 15;
    const int koff = (lane >> 4) << 1;           // 0 or 2
    const int n0   = blockIdx.x * 64;
    const int m0   = blockIdx.y * 128;

    v8f acc[4];
#pragma unroll
    for (int nt = 0; nt < 4; ++nt) {
        float cinit = use_bias ? Bm[(size_t)(n0 + nt * 16 + ml) * ldb] : 0.0f;
#pragma unroll
        for (int v = 0; v < 8; ++v) acc[nt][v] = cinit;
    }

    for (int k0 = 0; k0 < K; k0 += 32) {
        __syncthreads();
#pragma unroll
        for (int j = 0; j < 4; ++j) {            // stage A: 128x32 via float4
            int idx = tid + 256 * j;
            int row = idx >> 3;
            int c4  = (idx & 7) << 2;
            float4 av = *(const float4*)(A + (size_t)(m0 + row) * lda + k0 + c4);
            *(float4*)&sA[row * 36 + c4] = av;
            // pull next K-chunk toward the WGP while we compute this one
            if (k0 + 32 < K)
                __builtin_prefetch(A + (size_t)(m0 + row) * lda + k0 + 32 + c4, 0, 0);
        }
#pragma unroll
        for (int j = 0; j < 8; ++j) {            // stage B^T tile: 64x32 scalar
            int idx = tid + 256 * j;
            int row = idx >> 5;
            int cc  = idx & 31;
            sB[row * 36 + cc] = Bm[(size_t)(n0 + row) * ldb + kcol0 + k0 + cc];
        }
        __syncthreads();
#pragma unroll
        for (int kk = 0; kk < 32; kk += 4) {
            v2f a;
            a.x = sA[(wv * 16 + ml) * 36 + kk + koff];
            a.y = sA[(wv * 16 + ml) * 36 + kk + koff + 1];
#pragma unroll
            for (int nt = 0; nt < 4; ++nt) {
                v2f b;
                b.x = sB[(nt * 16 + ml) * 36 + kk + koff];
                b.y = sB[(nt * 16 + ml) * 36 + kk + koff + 1];
                acc[nt] = __builtin_amdgcn_wmma_f32_16x16x4_f32(
                              false, a, false, b, (short)0, acc[nt], false, false);
            }
        }
    }
    const int mrow = m0 + wv * 16 + ((lane >> 4) << 3);
#pragma unroll
    for (int nt = 0; nt < 4; ++nt)
#pragma unroll
        for (int v = 0; v < 8; ++v)
            C[(size_t)(mrow + v) * ldc + n0 + nt * 16 + ml] = acc[nt][v];
}

// ------------------------- deterministic CSR build -------------------------
__global__ void k_zero_i(int* p, int n) {
    int i = blockIdx.x * blockDim.x + threadIdx.x;
    if (i < n) p[i] = 0;
}
__global__ void k_hist(const int* __restrict__ row, int nnz, int ch,
                       int* __restrict__ counts) {
    int i = blockIdx.x * blockDim.x + threadIdx.x;
    if (i < nnz) atomicAdd(&counts[(i / ch) * RES + row[i]], 1);
}
__global__ void k_colscan(int* __restrict__ counts, int* __restrict__ row_tot) {
    int r = blockIdx.x * blockDim.x + threadIdx.x;
    if (r < RES) {
        int base = 0;
        for (int c = 0; c < NCHUNK; ++c) {
            int t = counts[c * RES + r];
            counts[c * RES + r] = base;
            base += t;
        }
        row_tot[r] = base;
    }
}
__global__ void k_rowscan(int* __restrict__ row_ptr) {
    if (threadIdx.x == 0 && blockIdx.x == 0) {
        int acc = 0;
        for (int r = 0; r < RES; ++r) { int t = row_ptr[r]; row_ptr[r] = acc; acc += t; }
        row_ptr[RES] = acc;
    }
}
__global__ void k_scatter(const int* __restrict__ row, const int* __restrict__ col,
                          const float* __restrict__ val, int nnz, int ch,
                          const int* __restrict__ row_ptr, int* __restrict__ counts,
                          int* __restrict__ csr_col, float* __restrict__ csr_val) {
    int c  = threadIdx.x;                        // one block of NCHUNK threads
    int i0 = c * ch;
    int i1 = i0 + ch; if (i1 > nnz) i1 = nnz;
    for (int i = i0; i < i1; ++i) {              // index order => deterministic
        int r   = row[i];
        int pos = row_ptr[r] + counts[c * RES + r];
        counts[c * RES + r] += 1;
        csr_col[pos] = col[i];
        csr_val[pos] = val[i];
    }
}
__global__ void k_copy_f(float* __restrict__ dst, const float* __restrict__ src, int n) {
    int i = blockIdx.x * blockDim.x + threadIdx.x;
    if (i < n) dst[i] = src[i];
}

// ----------------- ESN step: pre = WinU + W_sparse @ state ------------------
__global__ __launch_bounds__(256) void esn_step(
    const float* __restrict__ winu,      // [BATCH][RES] rows for this t
    const int* __restrict__ row_ptr,
    const int* __restrict__ csr_col,
    const float* __restrict__ csr_val,
    const float* __restrict__ s_cur,     // [BATCH][RES]
    float* __restrict__ s_nxt)
{
    __shared__ float st[RES];
    const int b   = blockIdx.y;
    const int seg = blockIdx.x;          // 0..7
    const int tid = threadIdx.x;
    for (int i = tid; i < RES; i += 256) st[i] = s_cur[(size_t)b * RES + i];
    __syncthreads();
    const int rbase = seg * (RES / 8);
    for (int r = rbase + tid; r < rbase + RES / 8; r += 256) {
        float acc = winu[(size_t)b * RES + r];
        int e1 = row_ptr[r + 1];
        for (int e = row_ptr[r]; e < e1; ++e)
            acc = fmaf(csr_val[e], st[csr_col[e]], acc);
        s_nxt[(size_t)b * RES + r] = (1.0f - ALEAK) * st[r] + ALEAK * tanhf(acc);
    }
}

// --------- attention step: softmax(Q.Ek) weighted by state, times Ev --------
__global__ __launch_bounds__(256) void attn_step(
    const float* __restrict__ Qt,        // [BATCH][EDIM] for this t
    const float* __restrict__ state,     // [BATCH][RES]  (new state)
    const float* __restrict__ Ek,        // [RES][HEADS][DHD]
    const float* __restrict__ Ev,
    float* __restrict__ outt)            // [BATCH][EDIM] for this t
{
    __shared__ float sS[RES];
    __shared__ float sSt[RES];
    __shared__ float sQ[DHD];
    __shared__ float red[256];
    const int h = blockIdx.x, b = blockIdx.y, tid = threadIdx.x;
    if (tid < DHD) sQ[tid] = Qt[(size_t)b * EDIM + h * DHD + tid];
    for (int i = tid; i < RES; i += 256) sSt[i] = state[(size_t)b * RES + i];
    __syncthreads();

    const float scale = 0.125f;          // 1/sqrt(64)
    float lmax = -1e30f;
    for (int i = 0; i < 16; ++i) {
        int n = tid + 256 * i;
        const float* ek = Ek + ((size_t)n * HEADS + h) * DHD;
        float d = 0.0f;
#pragma unroll 8
        for (int j = 0; j < DHD; ++j) d = fmaf(sQ[j], ek[j], d);
        d *= scale;
        sS[n] = d;
        lmax = fmaxf(lmax, d);
    }
    red[tid] = lmax; __syncthreads();
    for (int s = 128; s > 0; s >>= 1) {
        if (tid < s) red[tid] = fmaxf(red[tid], red[tid + s]);
        __syncthreads();
    }
    const float m = red[0];
    __syncthreads();

    float lsum = 0.0f;
    for (int i = 0; i < 16; ++i) {
        int n = tid + 256 * i;
        float e = __expf(sS[n] - m);
        sS[n] = e; lsum += e;
    }
    red[tid] = lsum; __syncthreads();
    for (int s = 128; s > 0; s >>= 1) {
        if (tid < s) red[tid] += red[tid + s];
        __syncthreads();
    }
    const float inv = 1.0f / red[0];
    __syncthreads();

    for (int i = 0; i < 16; ++i) {
        int n = tid + 256 * i;
        sS[n] = sS[n] * inv * sSt[n];    // attn * state
    }
    __syncthreads();

    const int d = tid & 63, seg = tid >> 6;
    float acc = 0.0f;
    for (int n = seg * (RES / 4); n < (seg + 1) * (RES / 4); ++n)
        acc = fmaf(sS[n], Ev[((size_t)n * HEADS + h) * DHD + d], acc);
    red[tid] = acc; __syncthreads();
    if (tid < 64)
        outt[(size_t)b * EDIM + h * DHD + tid] =
            red[tid] + red[tid + 64] + red[tid + 128] + red[tid + 192];
}

// ---------------------------------------------------------------------------
extern "C" void kernel_launch(void* const* d_in, const int* in_sizes, int n_in,
                              void* d_out, int out_size, void* d_ws, size_t ws_size,
                              hipStream_t stream) {
    const float* q    = (const float*)d_in[0];   // (S,B,E)
    const float* st0  = (const float*)d_in[1];   // (B,RES,1)
    const float* Win  = (const float*)d_in[2];   // (RES, 1+E)
    const float* Wq   = (const float*)d_in[3];   // (E,E)
    const float* Ek   = (const float*)d_in[4];   // (RES,H,DH)
    const float* Ev   = (const float*)d_in[5];   // (RES,H,DH)
    const float* Wval = (const float*)d_in[6];
    const int*   Wrow = (const int*)d_in[7];
    const int*   Wcol = (const int*)d_in[8];
    const int nnz = in_sizes[6];
    float* out = (float*)d_out;

    char* ws = (char*)d_ws;
    size_t off = 0;
    auto wsa = [&](size_t bytes) -> char* {
        char* p = ws + off;
        off = (off + bytes + 255) & ~(size_t)255;
        return p;
    };
    float* Qb      = (float*)wsa((size_t)SEQ * BATCH * EDIM * 4); // 32 MB
    float* WinU    = (float*)wsa((size_t)TCH * BATCH * RES * 4);  // 32 MB
    float* stA     = (float*)wsa((size_t)BATCH * RES * 4);
    float* stB     = (float*)wsa((size_t)BATCH * RES * 4);
    int*   counts  = (int*)wsa((size_t)NCHUNK * RES * 4);         // 4 MB
    int*   row_ptr = (int*)wsa((size_t)(RES + 1) * 4);
    float* csr_val = (float*)wsa((size_t)nnz * 4);
    int*   csr_col = (int*)wsa((size_t)nnz * 4);
    (void)ws_size; (void)out_size; (void)n_in;

    // COO -> CSR (deterministic chunked counting sort)
    const int ch = (nnz + NCHUNK - 1) / NCHUNK;
    k_zero_i<<<(NCHUNK * RES + 255) / 256, 256, 0, stream>>>(counts, NCHUNK * RES);
    k_hist<<<(nnz + 255) / 256, 256, 0, stream>>>(Wrow, nnz, ch, counts);
    k_colscan<<<(RES + 255) / 256, 256, 0, stream>>>(counts, row_ptr);
    k_rowscan<<<1, 1, 0, stream>>>(row_ptr);
    k_scatter<<<1, NCHUNK, 0, stream>>>(Wrow, Wcol, Wval, nnz, ch,
                                        row_ptr, counts, csr_col, csr_val);

    // init state
    k_copy_f<<<(BATCH * RES + 255) / 256, 256, 0, stream>>>(stA, st0, BATCH * RES);

    // Q = q @ Wq^T for all timesteps (16384 x 512 x 512)
    gemm_bt_wmma<<<dim3(EDIM / 64, (SEQ * BATCH) / 128), 256, 0, stream>>>(
        q, EDIM, Wq, EDIM, 0, 0, Qb, EDIM, EDIM);

    float* cur = stA; float* nxt = stB;
    for (int tc = 0; tc < SEQ / TCH; ++tc) {
        // WinU chunk = [1,q] @ Win^T  (2048 x 4096 x 513, bias = Win[:,0])
        gemm_bt_wmma<<<dim3(RES / 64, (TCH * BATCH) / 128), 256, 0, stream>>>(
            q + (size_t)tc * TCH * BATCH * EDIM, EDIM,
            Win, 1 + EDIM, /*kcol0=*/1, /*bias=*/1, WinU, RES, EDIM);
        for (int tt = 0; tt < TCH; ++tt) {
            int t = tc * TCH + tt;
            esn_step<<<dim3(8, BATCH), 256, 0, stream>>>(
                WinU + (size_t)tt * BATCH * RES, row_ptr, csr_col, csr_val, cur, nxt);
            float* tmp = cur; cur = nxt; nxt = tmp;   // cur = new state
            attn_step<<<dim3(HEADS, BATCH), 256, 0, stream>>>(
                Qb + (size_t)t * BATCH * EDIM, cur, Ek, Ev,
                out + (size_t)t * BATCH * EDIM);
        }
    }
    // final_state output
    k_copy_f<<<(BATCH * RES + 255) / 256, 256, 0, stream>>>(
        out + (size_t)SEQ * BATCH * EDIM, cur, BATCH * RES);
}